// Model_39676907882532
// MI455X (gfx1250) — compile-verified
//
#include <hip/hip_runtime.h>

// ---------------------------------------------------------------------------
// out[h,s,d] = qt[h,s,d] + a4[h,s] * biasb[h]
//   a4[h,s]  = jax threefry-bernoulli(key(42), keep)[h,s] ? 1/keep : 0
// (softmax over a singleton axis == 1.0, so qt@k^T and scaleb are dead code)
//
// Memory-bound: 424MB in + 424MB out -> ~36us floor at 23.3 TB/s.
// The (H,S,1)@(H,1,1)+qt step runs on the matrix pipe with
// V_WMMA_F32_16X16X4_F32 (D = A x B + C, C = qt tile loaded directly in the
// documented C/D VGPR layout -> every b32 access is two 64B segments).
// The stream has zero reuse and exceeds the 192MB L2, so all loads/stores
// carry non-temporal cache hints (TH=NT).
// ---------------------------------------------------------------------------

typedef __attribute__((ext_vector_type(2))) float v2f;
typedef __attribute__((ext_vector_type(8))) float v8f;

constexpr int H = 25920;
constexpr int S = 64;
constexpr int D = 64;
constexpr unsigned NMASK = (unsigned)H * (unsigned)S;   // 1,658,880 mask bits
constexpr unsigned HALFN = NMASK / 2;                   // threefry count split

constexpr float KEEP     = 0.2021470392102155f;         // 1 - P_DROP (as f32)
constexpr float INV_KEEP = 1.0f / KEEP;

__device__ __forceinline__ unsigned rotl32(unsigned x, int r) {
    return (x << r) | (x >> (32 - r));
}

// Exact JAX threefry2x32 (key = (0, 42) from jax.random.key(42))
__device__ __forceinline__ void threefry2x32(unsigned k0, unsigned k1,
                                             unsigned& x0, unsigned& x1) {
    const unsigned ks0 = k0, ks1 = k1, ks2 = k0 ^ k1 ^ 0x1BD11BDAu;
    const int ra[4] = {13, 15, 26, 6};
    const int rb[4] = {17, 29, 16, 24};
    x0 += ks0; x1 += ks1;
#pragma unroll
    for (int j = 0; j < 4; ++j) { x0 += x1; x1 = rotl32(x1, ra[j]); x1 ^= x0; }
    x0 += ks1; x1 += ks2 + 1u;
#pragma unroll
    for (int j = 0; j < 4; ++j) { x0 += x1; x1 = rotl32(x1, rb[j]); x1 ^= x0; }
    x0 += ks2; x1 += ks0 + 2u;
#pragma unroll
    for (int j = 0; j < 4; ++j) { x0 += x1; x1 = rotl32(x1, ra[j]); x1 ^= x0; }
    x0 += ks0; x1 += ks1 + 3u;
#pragma unroll
    for (int j = 0; j < 4; ++j) { x0 += x1; x1 = rotl32(x1, rb[j]); x1 ^= x0; }
    x0 += ks1; x1 += ks2 + 4u;
#pragma unroll
    for (int j = 0; j < 4; ++j) { x0 += x1; x1 = rotl32(x1, ra[j]); x1 ^= x0; }
    x0 += ks2; x1 += ks0 + 5u;
}

// One wave32 owns a 16(s) x 64(d) slab of one head: 4 WMMA 16x16 tiles.
// 8 waves per block -> grid = H*4/8 blocks.
__global__ __launch_bounds__(256) void
wmma_dropout_residual(const float* __restrict__ qt,
                      const float* __restrict__ biasb,
                      float* __restrict__ out) {
    const int lane = (int)(threadIdx.x & 31u);
    const int wave = (int)(threadIdx.x >> 5u);
    const int slab = (int)blockIdx.x * 8 + wave;   // [0, H*4)
    const int h  = slab >> 2;
    const int s0 = (slab & 3) << 4;

    // ---- dropout mask for rows s0..s0+15 (one threefry per lane, no branches)
    const unsigned i    = (unsigned)(h * S + s0) + (unsigned)(lane & 15);
    const unsigned pair = (i < HALFN) ? i : (i - HALFN);
    unsigned x0 = pair, x1 = pair + HALFN;
    threefry2x32(0u, 42u, x0, x1);
    const unsigned bits = (i < HALFN) ? x0 : x1;
    const float u  = __uint_as_float((bits >> 9) | 0x3F800000u) - 1.0f;
    const float a4 = (u < KEEP) ? INV_KEEP : 0.0f;

    // ---- A (16x4 f32): column K=0 = a4[row], all other K columns zero.
    // VGPR0: lanes0-15 hold K=0 (rows), lanes16-31 hold K=2 (must be 0).
    // VGPR1: K=1 / K=3 (must be 0).
    v2f A;
    A.x = (lane < 16) ? a4 : 0.0f;
    A.y = 0.0f;

    // ---- B (4x16 f32): only row K=0 contributes (A[:,1..3]==0), so a flat
    // broadcast of biasb[h] is exact regardless of B's row->lane mapping.
    const float bh = biasb[h];
    v2f B;
    B.x = bh;
    B.y = bh;

    // ---- C/D layout addressing: lane L (0-15) covers rows r, col L;
    // lanes 16-31 cover rows r+8, col L-16. Each b32 access = 2x64B segments.
    const long rowbase = ((long)h * S + s0 + ((lane >> 4) << 3)) * (long)D
                         + (long)(lane & 15);
    const float* qbase = qt  + rowbase;
    float*       obase = out + rowbase;

#pragma unroll
    for (int t = 0; t < 4; ++t) {           // 4 tiles across d
        v8f c;
#pragma unroll
        for (int r = 0; r < 8; ++r)
            c[r] = __builtin_nontemporal_load(qbase + t * 16 + r * D);

        // D = A x B + C  (8 args: neg_a, A, neg_b, B, c_mod, C, reuse_a, reuse_b)
        v8f d = __builtin_amdgcn_wmma_f32_16x16x4_f32(
            false, A, false, B, (short)0, c, false, false);

#pragma unroll
        for (int r = 0; r < 8; ++r)
            __builtin_nontemporal_store(d[r], obase + t * 16 + r * D);
    }
}

extern "C" void kernel_launch(void* const* d_in, const int* in_sizes, int n_in,
                              void* d_out, int out_size, void* d_ws, size_t ws_size,
                              hipStream_t stream) {
    // setup_inputs order: qt, k, scaleb, biasb. k/scaleb are dead (softmax over
    // a singleton axis is identically 1.0).
    const float* qt    = (const float*)d_in[0];
    const float* biasb = (const float*)d_in[3];
    float*       out   = (float*)d_out;

    const int slabs  = H * (S / 16);       // 103,680 wave-slabs
    const int blocks = slabs / 8;          // 8 waves (256 threads) per block
    hipLaunchKernelGGL(wmma_dropout_residual, dim3(blocks), dim3(256), 0, stream,
                       qt, biasb, out);
}